// GATReduce_37495064494699
// MI455X (gfx1250) — compile-verified
//
#include <hip/hip_runtime.h>
#include <math.h>

#define H 4
#define D 64
#define ROW (H * D)              // 256 floats = 1024 bytes per edge row
#define WAVES_PER_BLOCK 8
#define K2_BLOCK (WAVES_PER_BLOCK * 32)
#define K2_GRID 512

// gfx12 CPOL: TH[2:0] -> 0=RT, 1=NT (non-temporal streaming)
#define CPOL_NT 1

// ---------- order-preserving float<->uint key for atomic max ----------
__device__ __forceinline__ unsigned fkey(float f) {
  unsigned u = __float_as_uint(f);
  return (u & 0x80000000u) ? ~u : (u | 0x80000000u);
}
__device__ __forceinline__ float fkey_dec(unsigned k) {
  return (k & 0x80000000u) ? __uint_as_float(k & 0x7FFFFFFFu)
                           : __uint_as_float(~k);
}

// ---------- relaxed agent-scope atomics (lower to global_atomic_*) ----------
__device__ __forceinline__ void atomAddF32(float* p, float v) {
  __hip_atomic_fetch_add(p, v, __ATOMIC_RELAXED, __HIP_MEMORY_SCOPE_AGENT);
}
__device__ __forceinline__ void atomMaxU32(unsigned* p, unsigned v) {
  __hip_atomic_fetch_max(p, v, __ATOMIC_RELAXED, __HIP_MEMORY_SCOPE_AGENT);
}

// ---------- gfx1250 async global->LDS copy (ASYNCcnt-tracked) ----------
typedef int v4i __attribute__((ext_vector_type(4)));
typedef __attribute__((address_space(1))) v4i* gptr_v4i;   // ptr to AS1 int4
typedef __attribute__((address_space(3))) v4i* lptr_v4i;   // ptr to AS3 int4

__device__ __forceinline__ void async_copy_b128_nt(const float* gsrc, float* ldst) {
#if __has_builtin(__builtin_amdgcn_global_load_async_to_lds_b128)
  gptr_v4i g = (gptr_v4i)gsrc;
  lptr_v4i l = (lptr_v4i)ldst;
  __builtin_amdgcn_global_load_async_to_lds_b128(g, l, 0, CPOL_NT);
#else
  unsigned loff = (unsigned)(__SIZE_TYPE__)(__attribute__((address_space(3))) void*)ldst;
  asm volatile("global_load_async_to_lds_b128 %0, %1, off th:TH_LOAD_NT"
               :: "v"(loff), "v"(gsrc) : "memory");
#endif
}

__device__ __forceinline__ void wait_async_le2() {
#if __has_builtin(__builtin_amdgcn_s_wait_asynccnt)
  __builtin_amdgcn_s_wait_asynccnt(2);
#else
  asm volatile("s_wait_asynccnt 0x2" ::: "memory");
#endif
  asm volatile("" ::: "memory");
}
__device__ __forceinline__ void wait_async_0() {
#if __has_builtin(__builtin_amdgcn_s_wait_asynccnt)
  __builtin_amdgcn_s_wait_asynccnt(0);
#else
  asm volatile("s_wait_asynccnt 0x0" ::: "memory");
#endif
  asm volatile("" ::: "memory");
}

// ---------- kernel 0: init accumulators ----------
__global__ void gat_init(float* __restrict__ out, float* __restrict__ denom,
                         unsigned* __restrict__ smax, int NHD, int NH) {
  int i = blockIdx.x * blockDim.x + threadIdx.x;
  if (i < NHD) out[i] = 0.0f;
  if (i < NH) { denom[i] = 0.0f; smax[i] = 0u; }  // key 0 == "empty segment"
}

// ---------- kernel 1: per-(edge,head) attention logit + segment max ----------
__global__ void gat_edge_max(const float* __restrict__ a1,
                             const float* __restrict__ a2,
                             const int* __restrict__ dst,
                             unsigned* __restrict__ smax, int EH) {
  int i = blockIdx.x * blockDim.x + threadIdx.x;
  if (i >= EH) return;
  int e = i >> 2;          // / H
  int h = i & 3;           // % H
  int n = dst[e];
  float v = a1[n * H + h] + a2[i];
  v = v > 0.0f ? v : 0.01f * v;        // leaky_relu, slope 0.01
  atomMaxU32(&smax[n * H + h], fkey(v));
}

// ---------- kernel 2: main pass — async-staged ft rows + atomic scatter ----------
__global__ __launch_bounds__(K2_BLOCK) void gat_edge_accum(
    const float* __restrict__ a1, const float* __restrict__ a2,
    const float* __restrict__ ft, const int* __restrict__ dst,
    const unsigned* __restrict__ smax, float* __restrict__ num,
    float* __restrict__ denom, int E) {
  __shared__ float lds[WAVES_PER_BLOCK][2][ROW];   // 16 KB: 2x1KB per wave
  const int lane = threadIdx.x & 31;
  const int w = threadIdx.x >> 5;
  const int gw = blockIdx.x * WAVES_PER_BLOCK + w;
  const int nW = gridDim.x * WAVES_PER_BLOCK;
  float* buf0 = &lds[w][0][0];
  float* buf1 = &lds[w][1][0];

  // prologue: prefetch first edge row (lane covers 32B = 8 floats)
  int e = gw;
  if (e < E) {
    const float* src = ft + (size_t)e * ROW + lane * 8;
    async_copy_b128_nt(src, buf0 + lane * 8);
    async_copy_b128_nt(src + 4, buf0 + lane * 8 + 4);
  }
  int parity = 0;
  for (; e < E; e += nW) {
    int en = e + nW;
    if (en < E) {  // prefetch next row into other buffer, then drain oldest pair
      const float* src = ft + (size_t)en * ROW + lane * 8;
      float* dl = (parity ? buf0 : buf1) + lane * 8;
      async_copy_b128_nt(src, dl);
      async_copy_b128_nt(src + 4, dl + 4);
      wait_async_le2();
    } else {
      wait_async_0();
    }

    // attention coefficient for this (edge, head); lanes 0-7 h=0, 8-15 h=1, ...
    int n = dst[e];
    int h = lane >> 3;
    float att = a1[n * H + h] + a2[e * H + h];
    att = att > 0.0f ? att : 0.01f * att;
    float mx = fkey_dec(smax[n * H + h]);       // segment nonempty here
    float ex = __expf(att - mx);

    const float* b = (parity ? buf1 : buf0) + lane * 8;
    float* outp = num + ((size_t)n * H + h) * D + (lane & 7) * 8;
#pragma unroll
    for (int j = 0; j < 8; ++j) atomAddF32(outp + j, ex * b[j]);
    if ((lane & 7) == 0) atomAddF32(&denom[n * H + h], ex);

    parity ^= 1;
  }
}

// ---------- kernel 3: finalize out = num / max(denom,->1) ----------
__global__ void gat_final(float* __restrict__ out,
                          const float* __restrict__ denom, int NHD) {
  int i = blockIdx.x * blockDim.x + threadIdx.x;
  if (i >= NHD) return;
  float den = denom[i >> 6];   // i / D -> (n*H + h), D == 64
  out[i] = out[i] / (den > 0.0f ? den : 1.0f);
}

extern "C" void kernel_launch(void* const* d_in, const int* in_sizes, int n_in,
                              void* d_out, int out_size, void* d_ws, size_t ws_size,
                              hipStream_t stream) {
  const float* a1 = (const float*)d_in[0];   // [N,H,1]
  const float* a2 = (const float*)d_in[1];   // [E,H,1]
  const float* ft = (const float*)d_in[2];   // [E,H,D]
  const int* dst  = (const int*)d_in[3];     // [E]
  float* out = (float*)d_out;                // [N,H,D]

  const int NH  = in_sizes[0];               // N*H
  const int E   = in_sizes[3];
  const int EH  = E * H;
  const int NHD = NH * D;                    // == out_size

  float* denom   = (float*)d_ws;                                  // NH floats
  unsigned* smax = (unsigned*)((char*)d_ws + (size_t)NH * 4);     // NH uints

  gat_init<<<(NHD + 255) / 256, 256, 0, stream>>>(out, denom, smax, NHD, NH);
  gat_edge_max<<<(EH + 255) / 256, 256, 0, stream>>>(a1, a2, dst, smax, EH);
  gat_edge_accum<<<K2_GRID, K2_BLOCK, 0, stream>>>(a1, a2, ft, dst, smax, out,
                                                   denom, E);
  gat_final<<<(NHD + 255) / 256, 256, 0, stream>>>(out, denom, NHD);
}